// GINLayer_60868276519673
// MI455X (gfx1250) — compile-verified
//
#include <hip/hip_runtime.h>

typedef __attribute__((ext_vector_type(2))) float v2f;
typedef __attribute__((ext_vector_type(8))) float v8f;

#define IN_F 64
#define HID 256
#define OUT_F 64

// ---------------- Kernel 1: h0 = (1 + eps) * x  (float4 vectorized) -------
__global__ void gin_scale_kernel(const float* __restrict__ x,
                                 const float* __restrict__ eps,
                                 float* __restrict__ h0, int total4) {
    int i = blockIdx.x * blockDim.x + threadIdx.x;
    if (i >= total4) return;
    float s = 1.0f + eps[0];
    float4 v = ((const float4*)x)[i];
    v.x *= s; v.y *= s; v.z *= s; v.w *= s;
    ((float4*)h0)[i] = v;
}

// ---------------- Kernel 2: edge scatter-add ------------------------------
// 16 threads per edge; thread j handles features [4j, 4j+4).
__global__ void gin_scatter_kernel(const float* __restrict__ x,
                                   const int* __restrict__ src,
                                   const int* __restrict__ dst,
                                   float* __restrict__ h0, int E) {
    int gid = blockIdx.x * blockDim.x + threadIdx.x;
    int e = gid >> 4;
    if (e >= E) return;
    int j = gid & 15;
    const float4 v = ((const float4*)x)[src[e] * (IN_F / 4) + j];
    float* b = h0 + (long long)dst[e] * IN_F + j * 4;
    atomicAdd(b + 0, v.x);
    atomicAdd(b + 1, v.y);
    atomicAdd(b + 2, v.z);
    atomicAdd(b + 3, v.w);
}

// ---------------- Kernel 3: fused MLP via fp32 WMMA -----------------------
// One wave (32 lanes) computes 16 rows through both GEMMs.
// A-frag (16x4 f32): lane L: M = L&15; VGPR0 holds K = 2*(L>>4), VGPR1 = K+1.
// B-frag (4x16 f32): lane L: N = L&15; VGPR0 holds K = 2*(L>>4), VGPR1 = K+1.
// C/D (16x16 f32):   VGPR v: M = v + 8*(L>>4), N = L&15.
__global__ void __launch_bounds__(32)
gin_mlp_kernel(const float* __restrict__ h0, const float* __restrict__ W1,
               const float* __restrict__ b1, const float* __restrict__ W2,
               const float* __restrict__ b2, float* __restrict__ out) {
    __shared__ float hid[16 * 18];  // 16x16 tile, stride 18 (8B-aligned f2 reads)

    const int lane = threadIdx.x;        // 0..31
    const int lo   = lane & 15;
    const int hi   = lane >> 4;          // 0 or 1
    const int hi2  = hi << 1;            // 0 or 2
    const int r0   = blockIdx.x * 16;    // tile row base (N = 3125*16 exact)

    // ---- load A fragments for GEMM1: rows r0..r0+15, K = 0..63 ----
    v2f a1[16];
    const float* arow = h0 + (long long)(r0 + lo) * IN_F + hi2;
#pragma unroll
    for (int kk = 0; kk < 16; ++kk)
        a1[kk] = *(const v2f*)(arow + kk * 4);

    // ---- output accumulators, seeded with b2 ----
    v8f o[4];
#pragma unroll
    for (int t = 0; t < 4; ++t) {
        float bv = b2[t * 16 + lo];
#pragma unroll
        for (int v = 0; v < 8; ++v) o[t][v] = bv;
    }

    for (int n = 0; n < 16; ++n) {       // hidden tiles (HID/16)
        // hidden accumulator seeded with b1
        float b1v = b1[n * 16 + lo];
        v8f h;
#pragma unroll
        for (int v = 0; v < 8; ++v) h[v] = b1v;

        // ---- GEMM1: h(16x16) += A(16x64) * W1(64 x [16n..16n+15]) ----
        const float* w1col = W1 + n * 16 + lo;
#pragma unroll
        for (int kk = 0; kk < 16; ++kk) {
            int ks = kk * 4 + hi2;
            v2f b;
            b.x = w1col[ks * HID];
            b.y = w1col[(ks + 1) * HID];
            h = __builtin_amdgcn_wmma_f32_16x16x4_f32(false, a1[kk], false, b,
                                                      (short)0, h, false, false);
        }

        // ---- stage hidden tile: C-layout -> LDS (row major, stride 18) ----
#pragma unroll
        for (int v = 0; v < 8; ++v) {
            int m = v + (hi << 3);
            hid[m * 18 + lo] = h[v];
        }
        __syncthreads();   // single-wave WG: barrier == NOP + DS ordering

        // ---- GEMM2 partial: o(16x64) += hid(16x16) * W2[16n..16n+15, :] ----
        const float* w2col = W2 + lo;
#pragma unroll
        for (int j = 0; j < 4; ++j) {
            v2f a2 = *(const v2f*)&hid[lo * 18 + j * 4 + hi2];
            int kg = n * 16 + j * 4 + hi2;
#pragma unroll
            for (int t = 0; t < 4; ++t) {
                v2f b;
                b.x = w2col[kg * OUT_F + t * 16];
                b.y = w2col[(kg + 1) * OUT_F + t * 16];
                o[t] = __builtin_amdgcn_wmma_f32_16x16x4_f32(
                    false, a2, false, b, (short)0, o[t], false, false);
            }
        }
        __syncthreads();
    }

    // ---- write 16x64 output tile ----
#pragma unroll
    for (int t = 0; t < 4; ++t) {
#pragma unroll
        for (int v = 0; v < 8; ++v) {
            int m = v + (hi << 3);
            out[(long long)(r0 + m) * OUT_F + t * 16 + lo] = o[t][v];
        }
    }
}

// ---------------- Launch ---------------------------------------------------
extern "C" void kernel_launch(void* const* d_in, const int* in_sizes, int n_in,
                              void* d_out, int out_size, void* d_ws, size_t ws_size,
                              hipStream_t stream) {
    const float* x   = (const float*)d_in[0];
    const float* W1  = (const float*)d_in[1];
    const float* b1  = (const float*)d_in[2];
    const float* W2  = (const float*)d_in[3];
    const float* b2  = (const float*)d_in[4];
    const float* eps = (const float*)d_in[5];
    const int*   src = (const int*)d_in[6];
    const int*   dst = (const int*)d_in[7];
    float* out = (float*)d_out;

    const int N = in_sizes[0] / IN_F;   // 50000
    const int E = in_sizes[6];          // 800000

    // d_out doubles as the h0 = agg + (1+eps)x buffer ([N,64] == out shape).
    int total4 = N * IN_F / 4;
    gin_scale_kernel<<<(total4 + 255) / 256, 256, 0, stream>>>(x, eps, out, total4);

    long long sthreads = (long long)E * 16;
    gin_scatter_kernel<<<(int)((sthreads + 255) / 256), 256, 0, stream>>>(
        x, src, dst, out, E);

    gin_mlp_kernel<<<N / 16, 32, 0, stream>>>(out, W1, b1, W2, b2, out);
}